// MultiHeadAttention_755914244781
// MI455X (gfx1250) — compile-verified
//
#include <hip/hip_runtime.h>
#include <hip/hip_bf16.h>

typedef __attribute__((ext_vector_type(2))) float v2f;
typedef __attribute__((ext_vector_type(8))) float v8f;

#define Bb 2
#define Ss 2048
#define Dd 1024
#define Hh 16
#define DKk 64

// ---------------------------------------------------------------------------
// Tiled GEMM with bias: C[M,N] = A[M,K] @ B[K,N] + bias[N]
// Block tile 64x128, BK=32, 256 threads (8 wave32s). Each wave computes a
// 32x32 macro-tile as 2x2 WMMA tiles (v_wmma_f32_16x16x4_f32), reusing each
// A/B fragment twice to halve LDS traffic per wmma. Global->LDS staging is
// software-pipelined: next k-block's loads issue before the compute loop.
// All tile dims divide M/N/K exactly for this problem, so no bounds checks.
// ---------------------------------------------------------------------------
#define BM 64
#define BN 128
#define BK 32
#define AS_STRIDE 33
#define BS_STRIDE 129

__global__ __launch_bounds__(256)
void gemm_bias_wmma(const float* __restrict__ A, const float* __restrict__ Bm,
                    const float* __restrict__ bias, float* __restrict__ C,
                    int M, int N, int K) {
  __shared__ float As[BM * AS_STRIDE];   // 64 x 32 (+pad)
  __shared__ float Bs[BK * BS_STRIDE];   // 32 x 128 (+pad)

  const int m0 = blockIdx.x * BM;
  const int n0 = blockIdx.y * BN;
  const int t = threadIdx.x;      // 0..255
  const int lane = t & 31;
  const int w = t >> 5;           // wave id 0..7
  const int l16 = lane & 15;
  const int half = lane >> 4;     // 0: lanes 0-15, 1: lanes 16-31
  const int wrow = (w >> 2) * 32; // 0 or 32
  const int wcol = (w & 3) * 32;  // 0,32,64,96
  const int kfrag = 2 * half;     // K offset pair selected by lane half

  // A tile: 64x32 floats, two float4 per thread.
  const int arow = t >> 3;            // 0..31 (+32 for second chunk)
  const int acol = (t & 7) * 4;       // 0..28
  // B tile: 32x128 floats, four float4 per thread.
  const int brow = t >> 5;            // 0..7 (+8/16/24 for later chunks)
  const int bcol = (t & 31) * 4;      // 0..124

  v8f acc[2][2] = {};

  const float* aP = A  + (size_t)(m0 + arow) * K + acol;
  const float* bP = Bm + (size_t)brow * N + n0 + bcol;
  const size_t aRowStep = (size_t)32 * K;
  const size_t bRowStep = (size_t)8 * N;

  // Stage k-block 0 into registers.
  float4 av0 = *(const float4*)(aP);
  float4 av1 = *(const float4*)(aP + aRowStep);
  float4 bv0 = *(const float4*)(bP);
  float4 bv1 = *(const float4*)(bP + bRowStep);
  float4 bv2 = *(const float4*)(bP + 2 * bRowStep);
  float4 bv3 = *(const float4*)(bP + 3 * bRowStep);

  for (int k0 = 0; k0 < K; k0 += BK) {
    __syncthreads();  // previous iteration's LDS reads complete
    As[arow * AS_STRIDE + acol + 0] = av0.x;
    As[arow * AS_STRIDE + acol + 1] = av0.y;
    As[arow * AS_STRIDE + acol + 2] = av0.z;
    As[arow * AS_STRIDE + acol + 3] = av0.w;
    As[(arow + 32) * AS_STRIDE + acol + 0] = av1.x;
    As[(arow + 32) * AS_STRIDE + acol + 1] = av1.y;
    As[(arow + 32) * AS_STRIDE + acol + 2] = av1.z;
    As[(arow + 32) * AS_STRIDE + acol + 3] = av1.w;
    Bs[brow * BS_STRIDE + bcol + 0] = bv0.x;
    Bs[brow * BS_STRIDE + bcol + 1] = bv0.y;
    Bs[brow * BS_STRIDE + bcol + 2] = bv0.z;
    Bs[brow * BS_STRIDE + bcol + 3] = bv0.w;
    Bs[(brow + 8) * BS_STRIDE + bcol + 0] = bv1.x;
    Bs[(brow + 8) * BS_STRIDE + bcol + 1] = bv1.y;
    Bs[(brow + 8) * BS_STRIDE + bcol + 2] = bv1.z;
    Bs[(brow + 8) * BS_STRIDE + bcol + 3] = bv1.w;
    Bs[(brow + 16) * BS_STRIDE + bcol + 0] = bv2.x;
    Bs[(brow + 16) * BS_STRIDE + bcol + 1] = bv2.y;
    Bs[(brow + 16) * BS_STRIDE + bcol + 2] = bv2.z;
    Bs[(brow + 16) * BS_STRIDE + bcol + 3] = bv2.w;
    Bs[(brow + 24) * BS_STRIDE + bcol + 0] = bv3.x;
    Bs[(brow + 24) * BS_STRIDE + bcol + 1] = bv3.y;
    Bs[(brow + 24) * BS_STRIDE + bcol + 2] = bv3.z;
    Bs[(brow + 24) * BS_STRIDE + bcol + 3] = bv3.w;
    __syncthreads();

    // Issue next k-block's global loads now so they overlap the compute
    // below (uniform branch: EXEC stays all-1s for the WMMAs).
    const bool more = (k0 + BK) < K;
    if (more) {
      const float* aN = aP + (k0 + BK);
      const float* bN = bP + (size_t)(k0 + BK) * N;
      av0 = *(const float4*)(aN);
      av1 = *(const float4*)(aN + aRowStep);
      bv0 = *(const float4*)(bN);
      bv1 = *(const float4*)(bN + bRowStep);
      bv2 = *(const float4*)(bN + 2 * bRowStep);
      bv3 = *(const float4*)(bN + 3 * bRowStep);
      if (k0 + 2 * BK < K) {
        __builtin_prefetch(aP + (k0 + 2 * BK), 0, 1);
        __builtin_prefetch(bP + (size_t)(k0 + 2 * BK) * N, 0, 1);
      }
    }

#pragma unroll
    for (int kk = 0; kk < BK; kk += 4) {
      v2f af[2], bf[2];
      // A 16x4 fragments: lane holds row M = l16; VGPR0/1 = K = kfrag, +1
      af[0][0] = As[(wrow + l16) * AS_STRIDE + kk + kfrag];
      af[0][1] = As[(wrow + l16) * AS_STRIDE + kk + kfrag + 1];
      af[1][0] = As[(wrow + 16 + l16) * AS_STRIDE + kk + kfrag];
      af[1][1] = As[(wrow + 16 + l16) * AS_STRIDE + kk + kfrag + 1];
      // B 4x16 fragments: lane holds col N = l16; rows K = kfrag, +1
      bf[0][0] = Bs[(kk + kfrag)     * BS_STRIDE + wcol + l16];
      bf[0][1] = Bs[(kk + kfrag + 1) * BS_STRIDE + wcol + l16];
      bf[1][0] = Bs[(kk + kfrag)     * BS_STRIDE + wcol + 16 + l16];
      bf[1][1] = Bs[(kk + kfrag + 1) * BS_STRIDE + wcol + 16 + l16];
#pragma unroll
      for (int i = 0; i < 2; ++i)
#pragma unroll
        for (int j = 0; j < 2; ++j)
          acc[i][j] = __builtin_amdgcn_wmma_f32_16x16x4_f32(
              false, af[i], false, bf[j], (short)0, acc[i][j], false, false);
    }
  }

  // Epilogue: C/D layout — VGPR r: lanes 0-15 -> M=r, lanes 16-31 -> M=r+8.
#pragma unroll
  for (int j = 0; j < 2; ++j) {
    const int col = n0 + wcol + 16 * j + l16;
    const float bias_v = bias[col];
#pragma unroll
    for (int i = 0; i < 2; ++i) {
#pragma unroll
      for (int r = 0; r < 8; ++r) {
        const int row = m0 + wrow + 16 * i + r + 8 * half;
        C[(size_t)row * N + col] = acc[i][j][r] + bias_v;
      }
    }
  }
}

// ---------------------------------------------------------------------------
// Per-token attention-over-heads. One wave32 per (b,s) token.
// scores[h,t] = sum_d qh[h,d]*kh[t,d] / 8 ; softmax over t; ctx = probs @ vh.
// ---------------------------------------------------------------------------
__global__ __launch_bounds__(32)
void attn_wmma(const float* __restrict__ qp, const float* __restrict__ kp,
               const float* __restrict__ vp, float* __restrict__ probs_out,
               float* __restrict__ ctx_out) {
  __shared__ float Ps[16 * 17];

  const int token = blockIdx.x;        // b*S + s, 0..4095
  const int lane = threadIdx.x;        // 0..31
  const int l16 = lane & 15;
  const int half = lane >> 4;
  const int kfrag = 2 * half;

  const float* qb = qp + (size_t)token * Dd;
  const float* kb = kp + (size_t)token * Dd;
  const float* vb = vp + (size_t)token * Dd;

  // ---- scores = Q[16x64] @ K^T[64x16] -------------------------------------
  v8f acc = {};
#pragma unroll
  for (int kk = 0; kk < DKk; kk += 4) {
    v2f af, bf;
    af[0] = qb[l16 * DKk + kk + kfrag];        // A row h=l16, K=d
    af[1] = qb[l16 * DKk + kk + kfrag + 1];
    bf[0] = kb[l16 * DKk + kk + kfrag];        // B[d, t=l16] = kh[t, d]
    bf[1] = kb[l16 * DKk + kk + kfrag + 1];
    acc = __builtin_amdgcn_wmma_f32_16x16x4_f32(
        false, af, false, bf, (short)0, acc, false, false);
  }

  // ---- softmax over t (t = lane%16 in C layout) ---------------------------
  // Reduce within each 16-lane half with xor butterflies (wave32 native).
  float p[8];
#pragma unroll
  for (int r = 0; r < 8; ++r) {
    float x = acc[r] * 0.125f;  // 1/sqrt(64)
    float mx = x;
#pragma unroll
    for (int mask = 1; mask < 16; mask <<= 1)
      mx = fmaxf(mx, __shfl_xor(mx, mask, 32));
    float e = __expf(x - mx);
    float sum = e;
#pragma unroll
    for (int mask = 1; mask < 16; mask <<= 1)
      sum += __shfl_xor(sum, mask, 32);
    p[r] = e / sum;
  }

  // Emit probs [B,S,H,H]: row h = r + 8*half, col t = l16.
#pragma unroll
  for (int r = 0; r < 8; ++r) {
    probs_out[(size_t)token * (Hh * Hh) + (r + 8 * half) * Hh + l16] = p[r];
    Ps[(r + 8 * half) * 17 + l16] = p[r];   // stage for re-fragmenting as A
  }
  __syncthreads();

  // ---- ctx = probs[16x16] @ vh[16x64] -------------------------------------
#pragma unroll
  for (int nt = 0; nt < 4; ++nt) {
    v8f c2 = {};
#pragma unroll
    for (int kk = 0; kk < 16; kk += 4) {
      v2f af, bf;
      af[0] = Ps[l16 * 17 + kk + kfrag];              // A row h=l16, K=t
      af[1] = Ps[l16 * 17 + kk + kfrag + 1];
      bf[0] = vb[(kk + kfrag)     * DKk + nt * 16 + l16];  // B[t, d]
      bf[1] = vb[(kk + kfrag + 1) * DKk + nt * 16 + l16];
      c2 = __builtin_amdgcn_wmma_f32_16x16x4_f32(
          false, af, false, bf, (short)0, c2, false, false);
    }
#pragma unroll
    for (int r = 0; r < 8; ++r) {
      const size_t row = (size_t)token * Hh + r + 8 * half;  // flat (b,s,h)
      ctx_out[row * DKk + nt * 16 + l16] = c2[r];
    }
  }
}

// ---------------------------------------------------------------------------
extern "C" void kernel_launch(void* const* d_in, const int* in_sizes, int n_in,
                              void* d_out, int out_size, void* d_ws, size_t ws_size,
                              hipStream_t stream) {
  const float* v  = (const float*)d_in[0];
  const float* k  = (const float*)d_in[1];
  const float* q  = (const float*)d_in[2];
  const float* Wq = (const float*)d_in[3];
  const float* bq = (const float*)d_in[4];
  const float* Wk = (const float*)d_in[5];
  const float* bk = (const float*)d_in[6];
  const float* Wv = (const float*)d_in[7];
  const float* bv = (const float*)d_in[8];
  const float* Wo = (const float*)d_in[9];
  const float* bo = (const float*)d_in[10];

  float* out   = (float*)d_out;                                 // [B,S,H,D]
  float* probs = out + (size_t)Bb * Ss * Hh * Dd;               // [B,S,H,H]

  const int M1 = Bb * Ss;          // 4096 token rows
  float* qp  = (float*)d_ws;                       // [4096,1024]
  float* kp  = qp  + (size_t)M1 * Dd;
  float* vp  = kp  + (size_t)M1 * Dd;
  float* ctx = vp  + (size_t)M1 * Dd;              // [65536,64]

  dim3 blk(256);
  dim3 g1(M1 / BM, Dd / BN);                       // 64 x 8
  gemm_bias_wmma<<<g1, blk, 0, stream>>>(q, Wq, bq, qp, M1, Dd, Dd);
  gemm_bias_wmma<<<g1, blk, 0, stream>>>(k, Wk, bk, kp, M1, Dd, Dd);
  gemm_bias_wmma<<<g1, blk, 0, stream>>>(v, Wv, bv, vp, M1, Dd, Dd);

  attn_wmma<<<dim3(M1), dim3(32), 0, stream>>>(qp, kp, vp, probs, ctx);

  const int M3 = M1 * Hh;                          // 65536 (b,s,h) rows
  dim3 g3(M3 / BM, Dd / BN);                       // 1024 x 8
  gemm_bias_wmma<<<g3, blk, 0, stream>>>(ctx, Wo, bo, out, M3, Dd, DKk);
}